// SetConvThroughTime_86251533238888
// MI455X (gfx1250) — compile-verified
//
#include <hip/hip_runtime.h>

typedef float v2f __attribute__((ext_vector_type(2)));
typedef float v8f __attribute__((ext_vector_type(8)));

#define NB_B   2
#define NN     2048
#define NT_T   8
#define DX_D   3
#define DZ_D   16
#define HH     64
#define WW     64
#define GPB    (NT_T * HH * WW)      /* 32768 grid points per batch */
#define TILES_PB (GPB / 16)          /* 2048 16-row tiles per batch */
#define WAVES_PER_BLOCK 8
#define BLOCKS_PB (TILES_PB / WAVES_PER_BLOCK) /* 256 */

// -0.5 * log2(e) : exp(-0.5*d) == exp2(FNEG * d)
#define FNEG (-0.72134752044448170368f)
// -2 * FNEG == log2(e), folded into the staged dot-product coefficients
#define L2E  ( 1.4426950408889634074f)

__global__ __launch_bounds__(256)
void setconv_xgrid_kernel(const float* __restrict__ tg, float* __restrict__ out)
{
    int idx = blockIdx.x * 256 + threadIdx.x;       // 0 .. NB_B*GPB-1
    int b  = idx >> 15;
    int g  = idx & (GPB - 1);
    int wi = g & 63, hi = (g >> 6) & 63, ti = g >> 12;
    float* o = out + (size_t)idx * DX_D;
    o[0] = tg[b * NT_T + ti];
    o[1] = -1.0f + (float)hi * (2.0f / 63.0f);
    o[2] = -1.0f + (float)wi * (2.0f / 63.0f);
}

__global__ __launch_bounds__(256)
void setconv_fused_kernel(const float* __restrict__ x,
                          const float* __restrict__ z,
                          const float* __restrict__ tg,
                          const float* __restrict__ lsp,
                          float* __restrict__ zg_out)
{
    // Full-batch staging: 32KB coeffs + 128KB z = 160KB of the 320KB WGP LDS.
    __shared__ float4 cS[NN];
    __shared__ float  zS[NN * DZ_D];

    // inverse lengthscales: 1 / (1e-5 + softplus(p)); p is tiny (3 scalars).
    float ils0 = 1.0f / (1e-5f + log1pf(__expf(lsp[0])));
    float ils1 = 1.0f / (1e-5f + log1pf(__expf(lsp[1])));
    float ils2 = 1.0f / (1e-5f + log1pf(__expf(lsp[2])));

    int b        = blockIdx.x / BLOCKS_PB;
    int tileBase = (blockIdx.x % BLOCKS_PB) * WAVES_PER_BLOCK;

    const float* xb = x + (size_t)b * NN * DX_D;
    const float* zb = z + (size_t)b * NN * DZ_D;

    // Stage folded RBF coefficients: (L2E*c, FNEG*||c||^2) with c = x/ls.
    for (int n = threadIdx.x; n < NN; n += 256) {
        float c0 = xb[n * 3 + 0] * ils0;
        float c1 = xb[n * 3 + 1] * ils1;
        float c2 = xb[n * 3 + 2] * ils2;
        float cn2f = FNEG * (c0 * c0 + c1 * c1 + c2 * c2);
        cS[n] = make_float4(L2E * c0, L2E * c1, L2E * c2, cn2f);
    }
    // Stage the whole z[b] (B-matrix source) as float4 copies.
    {
        const float4* zb4 = (const float4*)zb;
        float4*       zS4 = (float4*)zS;
        for (int i = threadIdx.x; i < NN * DZ_D / 4; i += 256)
            zS4[i] = zb4[i];
    }
    __syncthreads();

    int lane = threadIdx.x & 31;
    int wave = threadIdx.x >> 5;
    int m    = lane & 15;               // A row / B col / D col
    int kb   = (lane >> 4) << 1;        // K sub-offset {0,2} per half-wave

    int gtile = tileBase + wave;        // one 16-row tile per wave
    int g  = gtile * 16 + m;
    int wi = g & 63, hi = (g >> 6) & 63, ti = g >> 12;

    // Per-row scaled grid coordinate a = xg/ls and folded norm FNEG*||a||^2.
    float a0 = tg[b * NT_T + ti] * ils0;
    float a1 = (-1.0f + (float)hi * (2.0f / 63.0f)) * ils1;
    float a2 = (-1.0f + (float)wi * (2.0f / 63.0f)) * ils2;
    float an2f = FNEG * (a0 * a0 + a1 * a1 + a2 * a2);

    v8f acc = {};
    #pragma unroll 8
    for (int nb = 0; nb < NN; nb += 4) {
        int n0 = nb + kb;
        float4 cp0 = cS[n0];
        float4 cp1 = cS[n0 + 1];
        // exponent = FNEG*(||a||^2 + ||c||^2 - 2 a.c); all scales pre-folded.
        // Explicit FMA nests -> v_pk_fma_f32 pairs instead of mul/add chains.
        float e0 = __builtin_fmaf(a0, cp0.x,
                   __builtin_fmaf(a1, cp0.y,
                   __builtin_fmaf(a2, cp0.z, an2f + cp0.w)));
        float e1 = __builtin_fmaf(a0, cp1.x,
                   __builtin_fmaf(a1, cp1.y,
                   __builtin_fmaf(a2, cp1.z, an2f + cp1.w)));

        v2f A;  A.x  = __builtin_amdgcn_exp2f(e0);   // v_exp_f32
                A.y  = __builtin_amdgcn_exp2f(e1);
        v2f Bm; Bm.x = zS[n0 * DZ_D + m];
                Bm.y = zS[(n0 + 1) * DZ_D + m];

        // D(16x16,f32) += A(16x4,f32) x B(4x16,f32) -- full f32 WMMA
        acc = __builtin_amdgcn_wmma_f32_16x16x4_f32(
                  false, A, false, Bm, (short)0, acc, false, false);
    }

    // C/D layout: VGPR r -> row (r + (lane>>4)*8), col m.
    float* outp = zg_out
        + ((size_t)b * GPB + (size_t)gtile * 16 + (size_t)(lane >> 4) * 8) * DZ_D
        + m;
    #pragma unroll
    for (int r = 0; r < 8; ++r)
        outp[(size_t)r * DZ_D] = acc[r];
}

extern "C" void kernel_launch(void* const* d_in, const int* in_sizes, int n_in,
                              void* d_out, int out_size, void* d_ws, size_t ws_size,
                              hipStream_t stream) {
    (void)in_sizes; (void)n_in; (void)out_size; (void)d_ws; (void)ws_size;
    const float* x   = (const float*)d_in[0];   // [B, N, DX]
    const float* z   = (const float*)d_in[1];   // [B, N, DZ]
    const float* tgr = (const float*)d_in[2];   // [B, NT]
    const float* lsp = (const float*)d_in[3];   // [DX]

    float* xg_out = (float*)d_out;                              // [B,NT,H,W,3]
    float* zg_out = xg_out + (size_t)NB_B * GPB * DX_D;         // [B,NT,H,W,16]

    setconv_xgrid_kernel<<<(NB_B * GPB) / 256, 256, 0, stream>>>(tgr, xg_out);
    setconv_fused_kernel<<<NB_B * BLOCKS_PB, 256, 0, stream>>>(x, z, tgr, lsp, zg_out);
}